// LocalRelationDistillLoss_79980880986714
// MI455X (gfx1250) — compile-verified
//
#include <hip/hip_runtime.h>
#include <math.h>

// Problem constants (from reference setup_inputs)
#define B_   16
#define N_   1024
#define D_   768
#define K_   8
#define BETA_ 0.5f
#define EPS_  1e-8f

typedef __attribute__((ext_vector_type(2))) float v2f;
typedef __attribute__((ext_vector_type(8))) float v8f;

// ---------------------------------------------------------------------------
// Kernel 1: per-anchor 8-NN in 2D center space.
// One thread per (b,n). Batch centers (8KB) staged in LDS. Register-resident
// unrolled insertion sort; stable on ties (matches jax top_k ordering).
// ---------------------------------------------------------------------------
__global__ __launch_bounds__(256) void knn_kernel(const float* __restrict__ centers,
                                                  int* __restrict__ idx_out) {
  __shared__ float sc[N_ * 2];
  const int bid = blockIdx.x;        // 64 blocks; 4 blocks per batch (N_/256)
  const int b   = bid >> 2;
  const int n0  = (bid & 3) * 256;

  for (int i = threadIdx.x; i < N_ * 2; i += 256)
    sc[i] = centers[(size_t)b * N_ * 2 + i];
  __syncthreads();

  const int n  = n0 + threadIdx.x;
  const float cx = sc[n * 2 + 0], cy = sc[n * 2 + 1];

  float bd[K_];
  int   bi[K_];
#pragma unroll
  for (int j = 0; j < K_; ++j) { bd[j] = 3.4e38f; bi[j] = -1; }

  for (int m = 0; m < N_; ++m) {
    if (m == n) continue;            // reference drops self (d2==0) slot
    const float dx = sc[m * 2 + 0] - cx;
    const float dy = sc[m * 2 + 1] - cy;
    const float d2 = dx * dx + dy * dy;
    if (d2 < bd[K_ - 1]) {           // strict < keeps earlier index on ties
      bd[K_ - 1] = d2; bi[K_ - 1] = m;
#pragma unroll
      for (int s = K_ - 1; s >= 1; --s) {
        if (bd[s] < bd[s - 1]) {
          float td = bd[s]; bd[s] = bd[s - 1]; bd[s - 1] = td;
          int   ti = bi[s]; bi[s] = bi[s - 1]; bi[s - 1] = ti;
        }
      }
    }
  }
#pragma unroll
  for (int j = 0; j < K_; ++j)
    idx_out[((size_t)b * N_ + n) * K_ + j] = bi[j];
}

// ---------------------------------------------------------------------------
// Kernel 2: per-anchor Gram matrix via V_WMMA_F32_16X16X4_F32.
// One wave32 per anchor. X = [anchor, nb0..nb7, anchor x7] (16 x 768).
// G = X * X^T accumulated over 192 K=4 steps; A and B fragments coincide
// because the Gram is symmetric and the f32 A layout (lane<16: K pair k..k+1
// of row `lane`; lane>=16: K pair k+2..k+3 of row `lane-16`) matches B.
// G[0][0]=|a|^2, G[0][1+j]=a.nb_j, G[1+j][1+j]=|nb_j|^2; padding rows only
// affect G entries with row/col >= 9 which are never read.
// ---------------------------------------------------------------------------
__global__ __launch_bounds__(256) void relloss_kernel(const float* __restrict__ s_emb,
                                                      const float* __restrict__ t_emb,
                                                      const int* __restrict__ idx,
                                                      float* __restrict__ partials) {
  __shared__ float wsum[8];
  const int lane = threadIdx.x & 31;
  const int wid  = threadIdx.x >> 5;
  const int w    = blockIdx.x * 8 + wid;     // anchor id, 0..16383
  const int b    = w >> 10;
  const int n    = w & (N_ - 1);

  // Row assignment for this lane's fragment slice.
  const int r = lane & 15;                   // matrix row of X
  int nn = n;                                // rows 0 and 9..15 -> anchor
  if (r >= 1 && r <= K_) nn = idx[(size_t)w * K_ + (r - 1)];
  const int q = (lane >> 4) << 1;            // K sub-offset: 0 or 2
  const float* srow = s_emb + ((size_t)b * N_ + nn) * D_ + q;
  const float* trow = t_emb + ((size_t)b * N_ + nn) * D_ + q;

  v8f cs = {0.f, 0.f, 0.f, 0.f, 0.f, 0.f, 0.f, 0.f};
  v8f ct = {0.f, 0.f, 0.f, 0.f, 0.f, 0.f, 0.f, 0.f};

  for (int k = 0; k < D_; k += 4) {
    v2f as, at;
    as.x = srow[k]; as.y = srow[k + 1];      // merged to global_load_b64
    at.x = trow[k]; at.y = trow[k + 1];
    // (neg_a, A, neg_b, B, c_mod, C, reuse_a, reuse_b)
    cs = __builtin_amdgcn_wmma_f32_16x16x4_f32(false, as, false, as, (short)0, cs, false, false);
    ct = __builtin_amdgcn_wmma_f32_16x16x4_f32(false, at, false, at, (short)0, ct, false, false);
  }

  // Spill accumulator fragments to scalars for shuffle extraction.
  float csa[8], cta[8];
#pragma unroll
  for (int i = 0; i < 8; ++i) { csa[i] = cs[i]; cta[i] = ct[i]; }

  // C/D layout: VGPR i holds M=i (lanes 0-15, N=lane) and M=i+8 (lanes 16-31).
  const float sG00 = __shfl(csa[0], 0, 32);  // |anchor|^2 (student)
  const float tG00 = __shfl(cta[0], 0, 32);
  const float na_s = fmaxf(sqrtf(sG00), EPS_);
  const float na_t = fmaxf(sqrtf(tG00), EPS_);

  float acc = 0.0f;
#pragma unroll
  for (int j = 0; j < K_; ++j) {
    // dot: G[0][j+1] -> VGPR0, lane j+1
    const float sd = __shfl(csa[0], j + 1, 32);
    const float td = __shfl(cta[0], j + 1, 32);
    // norm^2: G[j+1][j+1] -> j<7: VGPR j+1, lane j+1 ; j==7 (M=8): VGPR0, lane 24
    const float sn2 = (j < 7) ? __shfl(csa[j + 1], j + 1, 32) : __shfl(csa[0], 24, 32);
    const float tn2 = (j < 7) ? __shfl(cta[j + 1], j + 1, 32) : __shfl(cta[0], 24, 32);

    const float cos_s = sd / (na_s * fmaxf(sqrtf(sn2), EPS_));
    const float cos_t = td / (na_t * fmaxf(sqrtf(tn2), EPS_));
    const float diff  = fabsf(cos_s - cos_t);
    acc += (diff < BETA_) ? (0.5f * diff * diff / BETA_) : (diff - 0.5f * BETA_);
  }

  if (lane == 0) wsum[wid] = acc;
  __syncthreads();
  if (threadIdx.x == 0) {
    float s = 0.0f;
#pragma unroll
    for (int i = 0; i < 8; ++i) s += wsum[i];   // fixed order -> deterministic
    partials[blockIdx.x] = s;
  }
}

// ---------------------------------------------------------------------------
// Kernel 3: deterministic final reduction (fixed order, double accumulator).
// ---------------------------------------------------------------------------
__global__ void finalize_kernel(const float* __restrict__ partials, int nblocks,
                                float* __restrict__ out) {
  if (threadIdx.x == 0 && blockIdx.x == 0) {
    double s = 0.0;
    for (int i = 0; i < nblocks; ++i) s += (double)partials[i];
    out[0] = (float)(s / (double)((size_t)B_ * N_ * K_));
  }
}

// ---------------------------------------------------------------------------
extern "C" void kernel_launch(void* const* d_in, const int* in_sizes, int n_in,
                              void* d_out, int out_size, void* d_ws, size_t ws_size,
                              hipStream_t stream) {
  (void)in_sizes; (void)n_in; (void)out_size; (void)ws_size;
  const float* s_emb   = (const float*)d_in[0];   // (B,N,D) f32
  const float* t_emb   = (const float*)d_in[1];   // (B,N,D) f32
  const float* centers = (const float*)d_in[2];   // (B,N,2) f32

  int*   idx      = (int*)d_ws;                                   // B*N*K ints (512 KB)
  float* partials = (float*)((char*)d_ws + (size_t)B_ * N_ * K_ * sizeof(int));

  knn_kernel<<<(B_ * N_) / 256, 256, 0, stream>>>(centers, idx);
  relloss_kernel<<<(B_ * N_) / 8, 256, 0, stream>>>(s_emb, t_emb, idx, partials);
  finalize_kernel<<<1, 32, 0, stream>>>(partials, (B_ * N_) / 8, (float*)d_out);
}